// PiecewiseNDModel_54906861912559
// MI455X (gfx1250) — compile-verified
//
#include <hip/hip_runtime.h>
#include <stdint.h>

#define NBP 64

typedef float v4f __attribute__((ext_vector_type(4)));

// Branchless rank search. LDS table t holds {bp[1..63]} (t[63] unused/never read).
// Returns 4 * clamp(sum(x >= bp) - 1, 0, 63), i.e. the bucket index BYTE-scaled,
// so the DS vaddr is used directly and no re-scaling shifts are needed.
// Levels 1 and 2 of the probe tree come from scalar-loaded registers:
//   m = bp[32] (= t[31]), a = bp[16] (= t[15]), b = bp[48] (= t[47]).
__device__ __forceinline__ unsigned bucket64b(const float* __restrict__ t,
                                              float m, float a, float b, float x) {
    const bool c0 = (m <= x);
    unsigned posb = c0 ? 128u : 0u;
    const float v1 = c0 ? b : a;          // level-2 probe from registers (v_cndmask)
    posb += (v1 <= x) ? 64u : 0u;
#pragma unroll
    for (int half = 8; half >= 1; half >>= 1) {
        // ds_load_b32: vaddr = posb, immediate offset = dim_base + (half-1)*4
        const float v = *(const float*)((const char*)t + posb + (unsigned)((half - 1) * 4));
        posb += (v <= x) ? (unsigned)(half * 4) : 0u;  // conflict-free probe tree
    }
    return posb;  // = idx * 4
}

// p0,p1,p2 are byte-scaled indices (idx*4). Byte offset into values =
// idx0*16384 + idx1*256 + idx2*4 = (p0<<12)|(p1<<6)|p2 (disjoint bit fields).
__device__ __forceinline__ float gather3d(const float* __restrict__ values,
                                          unsigned p0, unsigned p1, unsigned p2) {
    const unsigned byte_off = (p0 << 12) | (p1 << 6) | p2;
    return *(const float*)((const char*)values + byte_off);   // L2-resident gather
}

__global__ __launch_bounds__(256) void piecewise3d_kernel(
    const float* __restrict__ x0, const float* __restrict__ x1, const float* __restrict__ x2,
    const float* __restrict__ bp0, const float* __restrict__ bp1, const float* __restrict__ bp2,
    const float* __restrict__ values, float* __restrict__ out, int n)
{
    __shared__ float t[3 * NBP];  // dim d at t[d*64 + i]; entry i -> LDS bank i (conflict-free)

    const int tid = threadIdx.x;

    // ---- Stage shifted breakpoints into LDS via CDNA5 async global->LDS ----
    // t[d*64 + i] = bp_d[i+1] for i < 63 ; t[d*64 + 63] is never read.
    if (tid < 3 * NBP && (tid & 63) < NBP - 1) {
        const int d = tid >> 6;
        const int i = tid & 63;
        const float* bp = (d == 0) ? bp0 : ((d == 1) ? bp1 : bp2);
        unsigned lds_off = (unsigned)(uintptr_t)&t[tid];          // low 32 bits = LDS byte offset
        unsigned long long gaddr = (unsigned long long)(uintptr_t)(bp + i);
        // GLOBAL_LOAD_ASYNC_TO_LDS_B32: vdst = LDS byte address, vaddr = 64-bit global addr.
        // offset:4 shifts the source to bp[i+1].
        asm volatile("global_load_async_to_lds_b32 %0, %1, off offset:4"
                     :: "v"(lds_off), "v"(gaddr)
                     : "memory");
    }

    // Probe-tree levels 1 and 2: uniform addresses -> scalar (SMEM) loads, no DS traffic.
    const float m0 = bp0[32], a0 = bp0[16], b0 = bp0[48];
    const float m1 = bp1[32], a1 = bp1[16], b1 = bp1[48];
    const float m2 = bp2[32], a2 = bp2[16], b2 = bp2[48];

    asm volatile("s_wait_asynccnt 0" ::: "memory");   // async transfers landed in LDS
    __syncthreads();

    const float* t0 = t;
    const float* t1 = t + NBP;       // byte offsets 256 / 512 fold into DS offset immediates
    const float* t2 = t + 2 * NBP;

    const long long base = ((long long)blockIdx.x * blockDim.x + tid) * 8;
    if (base >= n) return;

    if (base + 8 <= (long long)n) {
        // Streaming reads: non-temporal so the 1 MB values table stays resident in L2.
        const v4f va0 = __builtin_nontemporal_load((const v4f*)(x0 + base));
        const v4f va1 = __builtin_nontemporal_load((const v4f*)(x0 + base + 4));
        const v4f vb0 = __builtin_nontemporal_load((const v4f*)(x1 + base));
        const v4f vb1 = __builtin_nontemporal_load((const v4f*)(x1 + base + 4));
        const v4f vc0 = __builtin_nontemporal_load((const v4f*)(x2 + base));
        const v4f vc1 = __builtin_nontemporal_load((const v4f*)(x2 + base + 4));
        v4f r0, r1;
#pragma unroll
        for (int k = 0; k < 4; ++k) {
            r0[k] = gather3d(values,
                             bucket64b(t0, m0, a0, b0, va0[k]),
                             bucket64b(t1, m1, a1, b1, vb0[k]),
                             bucket64b(t2, m2, a2, b2, vc0[k]));
        }
#pragma unroll
        for (int k = 0; k < 4; ++k) {
            r1[k] = gather3d(values,
                             bucket64b(t0, m0, a0, b0, va1[k]),
                             bucket64b(t1, m1, a1, b1, vb1[k]),
                             bucket64b(t2, m2, a2, b2, vc1[k]));
        }
        __builtin_nontemporal_store(r0, (v4f*)(out + base));
        __builtin_nontemporal_store(r1, (v4f*)(out + base + 4));
    } else {
        // Generic tail (not hit for n = 8388608, but keep it correct for any n).
        for (long long k = base; k < (long long)n; ++k) {
            const unsigned p0 = bucket64b(t0, m0, a0, b0, x0[k]);
            const unsigned p1 = bucket64b(t1, m1, a1, b1, x1[k]);
            const unsigned p2 = bucket64b(t2, m2, a2, b2, x2[k]);
            out[k] = gather3d(values, p0, p1, p2);
        }
    }
}

extern "C" void kernel_launch(void* const* d_in, const int* in_sizes, int n_in,
                              void* d_out, int out_size, void* d_ws, size_t ws_size,
                              hipStream_t stream) {
    const float* x0     = (const float*)d_in[0];
    const float* x1     = (const float*)d_in[1];
    const float* x2     = (const float*)d_in[2];
    const float* bp0    = (const float*)d_in[3];
    const float* bp1    = (const float*)d_in[4];
    const float* bp2    = (const float*)d_in[5];
    const float* values = (const float*)d_in[6];
    float* out          = (float*)d_out;

    const int n = in_sizes[0];
    const int threads = 256;
    const int points_per_block = threads * 8;
    const int blocks = (int)((n + (long long)points_per_block - 1) / points_per_block);

    piecewise3d_kernel<<<blocks, threads, 0, stream>>>(x0, x1, x2, bp0, bp1, bp2, values, out, n);
}